// InferenceBox_12730283065472
// MI455X (gfx1250) — compile-verified
//
#include <hip/hip_runtime.h>
#include <stdint.h>

// Match XLA: mul and sub/add are separate HLOs, no FMA contraction.
#pragma clang fp contract(off)

#define N_BOX   32768
#define NCLS    81
#define NCLS_OUT 80
#define NBATCH  32
#define TOPK    200
#define CAND    1024
#define THREADS 256
#define NWAVES  8
#define ROWS_PW (N_BOX / NWAVES)   // 4096 rows per wave's TDM slice
#define CONF_THR 0.01f
#define IOU_THR  0.45f

typedef unsigned int uint;
typedef uint u32x4v __attribute__((ext_vector_type(4)));
typedef int  i32x4v __attribute__((ext_vector_type(4)));
typedef int  i32x8v __attribute__((ext_vector_type(8)));

__global__ __launch_bounds__(THREADS) void nms_fused_kernel(
    const float* __restrict__ loc,    // (B, N, 4) cx,cy,w,h
    const float* __restrict__ conf,   // (B, N, 81)
    float* __restrict__ out_boxes,    // (B, 80, 200, 6)
    float* __restrict__ out_keep)     // (B, 80, 200)
{
    __shared__ float s_conf[N_BOX];                 // 128 KB: whole conf column
    __shared__ unsigned long long s_key[CAND];      // 8 KB sort buffer
    __shared__ uint s_hist[256];
    __shared__ uint s_cnt;
    __shared__ uint s_b1, s_cumA, s_pref16;
    __shared__ float s_bx[TOPK][4];                 // raw centroid boxes
    __shared__ float s_x0[TOPK], s_y0[TOPK], s_x1[TOPK], s_y1[TOPK], s_area[TOPK];
    __shared__ float s_cf[TOPK];
    __shared__ uint s_mask[TOPK][7];                // suppression bitmask rows (j>i only)
    __shared__ uint s_keepw[7];

    const uint tid  = threadIdx.x;
    const uint pair = blockIdx.x;
    const uint b    = pair / NCLS_OUT;
    const uint c    = pair % NCLS_OUT;

    // ---------------- TDM: DMA conf[b, :, c] (stride 81 dwords) into LDS --------------
    // Each wave moves a 4096-row slice: 2D tile, tile_dim0 = 1 elem, tile_dim1 = 4096,
    // tensor_dim0_stride = 81 (data_size = 4B). Contiguous LDS destination.
    {
        const float* col = conf + ((size_t)b * N_BOX) * NCLS + c;
        uint wid = (uint)__builtin_amdgcn_readfirstlane((int)(tid >> 5));
        uint64_t ga = (uint64_t)(uintptr_t)(col + (size_t)wid * ROWS_PW * NCLS);
        // flat LDS address: low 32 bits are the wave-relative LDS byte offset
        uint lds_off = (uint)(uintptr_t)(void*)&s_conf[wid * ROWS_PW];

        u32x4v g0;
        g0.x = 1u;                                          // count=1, user descriptor
        g0.y = lds_off;                                     // lds_addr
        g0.z = (uint)ga;                                    // global_addr[31:0]
        g0.w = ((uint)(ga >> 32) & 0x01FFFFFFu) | 0x80000000u; // addr[56:32] | type=2

        i32x8v g1;
        g1[0] = (int)(2u << 16);            // workgroup_mask=0 | data_size=2 (4B)
        g1[1] = (int)((uint)NCLS << 16);    // atomic_barrier=0 | tensor_dim0[15:0]=81
        g1[2] = (int)((uint)ROWS_PW << 16); // tensor_dim0[31:16]=0 | tensor_dim1[15:0]
        g1[3] = (int)(1u << 16);            // tensor_dim1[31:16]=0 | tile_dim0=1
        g1[4] = (int)(uint)ROWS_PW;         // tile_dim1=4096 | tile_dim2=0
        g1[5] = (int)(uint)NCLS;            // tensor_dim0_stride[31:0]=81
        g1[6] = 0;                          // stride0 hi | dim1_stride lo (unused, 2D tile)
        g1[7] = 0;

        i32x4v gz4 = {0, 0, 0, 0};
        i32x8v gz8 = {0, 0, 0, 0, 0, 0, 0, 0};
        __builtin_amdgcn_tensor_load_to_lds(g0, g1, gz4, gz4, gz8, 0);
        __builtin_amdgcn_s_wait_tensorcnt(0);
    }
    __syncthreads();

    // ---------------- exact top-200: two-level radix histogram on float bits -----------
    for (uint i = tid; i < 256; i += THREADS) s_hist[i] = 0;
    if (tid == 0) s_cnt = 0;
    __syncthreads();

    for (uint n = tid; n < N_BOX; n += THREADS) {
        float v = s_conf[n];
        uint key = (v > CONF_THR) ? __float_as_uint(v) : 0u; // positive floats: bit order = value order
        atomicAdd(&s_hist[key >> 24], 1u);
    }
    __syncthreads();

    if (tid == 0) {
        uint cum = 0, b1 = 0;
        for (int bin = 255; bin >= 0; --bin) {
            uint h = s_hist[bin];
            if (cum + h >= TOPK) { b1 = (uint)bin; break; }
            cum += h;
        }
        s_b1 = b1; s_cumA = cum;
    }
    __syncthreads();
    const uint b1 = s_b1;
    for (uint i = tid; i < 256; i += THREADS) s_hist[i] = 0;
    __syncthreads();

    for (uint n = tid; n < N_BOX; n += THREADS) {
        float v = s_conf[n];
        uint key = (v > CONF_THR) ? __float_as_uint(v) : 0u;
        if ((key >> 24) == b1) atomicAdd(&s_hist[(key >> 16) & 0xFFu], 1u);
    }
    __syncthreads();

    if (tid == 0) {
        uint cum = s_cumA, b2 = 0;
        for (int bin = 255; bin >= 0; --bin) {
            uint h = s_hist[bin];
            if (cum + h >= TOPK) { b2 = (uint)bin; break; }
            cum += h;
        }
        s_pref16 = (b1 << 8) | b2;
    }
    __syncthreads();
    const uint pref16 = s_pref16;

    // collect candidates with (key>>16) >= pref16 (>= 200 of them when >=200 pass thr)
    for (uint n = tid; n < N_BOX; n += THREADS) {
        float v = s_conf[n];
        uint key = (v > CONF_THR) ? __float_as_uint(v) : 0u;
        if (key != 0u && (key >> 16) >= pref16) {
            uint pos = atomicAdd(&s_cnt, 1u);
            if (pos < CAND)
                s_key[pos] = ((unsigned long long)key << 32) | (uint)(~n); // ~n: lower idx wins ties
        }
    }
    __syncthreads();
    {
        uint cnt = s_cnt; if (cnt > CAND) cnt = CAND;
        for (uint i = tid; i < CAND; i += THREADS)
            if (i >= cnt) s_key[i] = 0ull;
    }

    // bitonic sort, descending, 1024 x u64 in LDS
    for (uint k = 2; k <= CAND; k <<= 1) {
        for (uint j = k >> 1; j > 0; j >>= 1) {
            __syncthreads();
            for (uint p = tid; p < (CAND / 2); p += THREADS) {
                uint i = 2u * p - (p & (j - 1u));
                uint l = i | j;
                unsigned long long a = s_key[i], bb = s_key[l];
                bool dirAsc = ((i & k) != 0u);   // inverted -> global descending
                if (dirAsc ? (a > bb) : (a < bb)) { s_key[i] = bb; s_key[l] = a; }
            }
        }
    }
    __syncthreads();

    // ---------------- gather boxes, corners, areas ------------------------------------
    for (uint i = tid; i < TOPK; i += THREADS) {
        unsigned long long key = s_key[i];
        uint vb = (uint)(key >> 32);
        float cf = __uint_as_float(vb);
        float b0 = 0.f, bb1 = 0.f, b2 = 0.f, b3 = 0.f;
        if (vb != 0u) {
            uint n = ~(uint)key;
            const float* lp = loc + (((size_t)b * N_BOX) + n) * 4;
            b0 = lp[0]; bb1 = lp[1]; b2 = lp[2]; b3 = lp[3];
        } else {
            cf = -1.0f;   // below-threshold slot: keep=false, output -1
        }
        s_cf[i] = cf;
        s_bx[i][0] = b0; s_bx[i][1] = bb1; s_bx[i][2] = b2; s_bx[i][3] = b3;
        float hw = b2 * 0.5f, hh = b3 * 0.5f;
        float x0 = b0 - hw, y0 = bb1 - hh, x1 = b0 + hw, y1 = bb1 + hh;
        s_x0[i] = x0; s_y0[i] = y0; s_x1[i] = x1; s_y1[i] = y1;
        s_area[i] = (x1 - x0) * (y1 - y0);
    }
    for (uint i = tid; i < TOPK * 7; i += THREADS) ((uint*)s_mask)[i] = 0u;
    __syncthreads();

    // ---------------- 200x200 IoU -> suppression bitmask (j > i only) ------------------
    for (uint t = tid; t < TOPK * TOPK; t += THREADS) {
        uint i = t / TOPK, j = t % TOPK;
        if (j <= i) continue;
        float lx = fmaxf(s_x0[i], s_x0[j]);
        float ly = fmaxf(s_y0[i], s_y0[j]);
        float rx = fminf(s_x1[i], s_x1[j]);
        float ry = fminf(s_y1[i], s_y1[j]);
        float w = fmaxf(rx - lx, 0.0f);
        float h = fmaxf(ry - ly, 0.0f);
        float inter = w * h;
        float iou = inter / (s_area[i] + s_area[j] - inter);
        if (iou > IOU_THR) atomicOr(&s_mask[i][j >> 5], 1u << (j & 31u));
    }
    __syncthreads();

    // ---------------- greedy NMS (sequential, cheap on bitmask words) ------------------
    if (tid == 0) {
        uint ku[7];
        for (int w = 0; w < 7; ++w) ku[w] = 0u;
        for (uint i = 0; i < TOPK; ++i)
            if (s_cf[i] > CONF_THR) ku[i >> 5] |= (1u << (i & 31u));
        for (uint i = 0; i < TOPK; ++i) {
            if (ku[i >> 5] & (1u << (i & 31u))) {
                for (int w = 0; w < 7; ++w) ku[w] &= ~s_mask[i][w]; // rows hold only j>i bits
            }
        }
        for (int w = 0; w < 7; ++w) s_keepw[w] = ku[w];
    }
    __syncthreads();

    // ---------------- outputs ----------------------------------------------------------
    for (uint i = tid; i < TOPK; i += THREADS) {
        bool kept = (s_keepw[i >> 5] >> (i & 31u)) & 1u;
        size_t base = ((size_t)pair * TOPK + i) * 6;
        if (kept) {
            out_boxes[base + 0] = (float)c;
            out_boxes[base + 1] = s_cf[i];
            out_boxes[base + 2] = s_bx[i][0];
            out_boxes[base + 3] = s_bx[i][1];
            out_boxes[base + 4] = s_bx[i][2];
            out_boxes[base + 5] = s_bx[i][3];
        } else {
            out_boxes[base + 0] = -1.0f;
            out_boxes[base + 1] = -1.0f;
            out_boxes[base + 2] = -1.0f;
            out_boxes[base + 3] = -1.0f;
            out_boxes[base + 4] = -1.0f;
            out_boxes[base + 5] = -1.0f;
        }
        out_keep[(size_t)pair * TOPK + i] = kept ? 1.0f : 0.0f;
    }
}

extern "C" void kernel_launch(void* const* d_in, const int* in_sizes, int n_in,
                              void* d_out, int out_size, void* d_ws, size_t ws_size,
                              hipStream_t stream) {
    const float* loc  = (const float*)d_in[0];   // (32, 32768, 4)
    const float* conf = (const float*)d_in[1];   // (32, 32768, 81)
    float* out_boxes = (float*)d_out;                                         // (32,80,200,6)
    float* out_keep  = (float*)d_out + (size_t)NBATCH * NCLS_OUT * TOPK * 6;  // (32,80,200)

    dim3 grid(NBATCH * NCLS_OUT);   // 2560 independent (batch, class) problems
    dim3 block(THREADS);            // 8 wave32 waves
    nms_fused_kernel<<<grid, block, 0, stream>>>(loc, conf, out_boxes, out_keep);
}